// CholeskyResHead_68255620268805
// MI455X (gfx1250) — compile-verified
//
#include <hip/hip_runtime.h>
#include <math.h>

// ---------------- problem constants ----------------
#define BB   64
#define NN   207
#define TT   12
#define CC   10
#define NP   208            // N padded to multiple of 16
#define NP2  (NP*NP)        // 43264
#define LDSP 209            // LDS pitch for Cholesky (bank-conflict-free)
#define NTOT (BB*NN*TT)     // 158976
#define LOG2PI 1.8378770664093453f

typedef float v2f __attribute__((ext_vector_type(2)));
typedef float v8f __attribute__((ext_vector_type(8)));

// ---------------- workspace layout (float offsets) ----------------
#define O_MAEP  0           // 621*2
#define O_UT    2048        // 10*144
#define O_DT    3488        // 120
#define O_ULOG  3616        // 10
#define O_VLOG  3632        // 10
#define O_KS    4096        // 10*NP2
#define O_G     444416      // 120*NP2
#define O_Z     5636096     // 120*NP*64
#define O_QUADP 7233536     // 120*64   (total ~28.97 MB)

// ============ 1) Kt = Lt Lt^T, 12x12 Jacobi eigen, log-dets ============
__global__ void kt_eigen_kernel(const float* __restrict__ Lt,
                                const float* __restrict__ Ls,
                                float* __restrict__ Ut, float* __restrict__ Dt,
                                float* __restrict__ Ulog, float* __restrict__ Vlog) {
  __shared__ float JA[CC][TT][TT];
  __shared__ float JV[CC][TT][TT];
  int c = threadIdx.x;
  if (c >= CC) return;
  const float* L = Lt + c * TT * TT;
  for (int i = 0; i < TT; ++i)
    for (int j = 0; j < TT; ++j) {
      float s = 0.f;
      for (int k = 0; k < TT; ++k) s += L[i*TT+k] * L[j*TT+k];
      JA[c][i][j] = s;
      JV[c][i][j] = (i == j) ? 1.f : 0.f;
    }
  for (int sweep = 0; sweep < 30; ++sweep) {
    for (int p = 0; p < TT - 1; ++p)
      for (int q = p + 1; q < TT; ++q) {
        float apq = JA[c][p][q];
        if (fabsf(apq) <= 1e-20f) continue;
        float app = JA[c][p][p], aqq = JA[c][q][q];
        float tau = (aqq - app) / (2.f * apq);
        float t   = ((tau >= 0.f) ? 1.f : -1.f) / (fabsf(tau) + sqrtf(1.f + tau*tau));
        float cc  = 1.f / sqrtf(1.f + t*t);
        float ss  = t * cc;
        for (int k = 0; k < TT; ++k) {        // A <- A*J (columns)
          float akp = JA[c][k][p], akq = JA[c][k][q];
          JA[c][k][p] = cc*akp - ss*akq;
          JA[c][k][q] = ss*akp + cc*akq;
        }
        for (int k = 0; k < TT; ++k) {        // A <- J^T*A (rows)
          float apk = JA[c][p][k], aqk = JA[c][q][k];
          JA[c][p][k] = cc*apk - ss*aqk;
          JA[c][q][k] = ss*apk + cc*aqk;
        }
        for (int k = 0; k < TT; ++k) {        // V <- V*J
          float vkp = JV[c][k][p], vkq = JV[c][k][q];
          JV[c][k][p] = cc*vkp - ss*vkq;
          JV[c][k][q] = ss*vkp + cc*vkq;
        }
      }
  }
  for (int j = 0; j < TT; ++j) Dt[c*TT + j] = JA[c][j][j];
  for (int t = 0; t < TT; ++t)
    for (int j = 0; j < TT; ++j) Ut[c*144 + t*TT + j] = JV[c][t][j];
  float vl = 0.f;
  for (int t = 0; t < TT; ++t) vl += logf(L[t*TT + t]);
  Vlog[c] = vl;
  const float* Sl = Ls + c * NN * NN;
  float ul = 0.f;
  for (int i = 0; i < NN; ++i) ul += logf(Sl[i*NN + i]);
  Ulog[c] = ul;
}

// ============ 2) Ks = Ls Ls^T via V_WMMA_F32_16X16X4_F32 (padded 208) ====
// Branchless: clamp indices into valid memory (unconditional global_load)
// and zero padded rows/cols via mask multiplies. Only the last K-chunk
// needs column masks, so the 50-chunk main loop is mask-free on K.
__global__ void ks_wmma_kernel(const float* __restrict__ Ls, float* __restrict__ Ks) {
  int c = blockIdx.y;
  int tile = blockIdx.x;            // 13*13 tiles
  int tm = tile / 13, tn = tile % 13;
  int lane = threadIdx.x & 31;
  bool hi = lane >= 16;
  int l = lane & 15;
  int row  = tm * 16 + l;           // A row (M)
  int coln = tn * 16 + l;           // B column (N)
  const float* L = Ls + c * NN * NN;
  int rowc  = row  < NN ? row  : NN - 1;
  int colnc = coln < NN ? coln : NN - 1;
  float rm = (row  < NN) ? 1.f : 0.f;
  float cm = (coln < NN) ? 1.f : 0.f;
  const float* Ar = L + rowc  * NN;
  const float* Br = L + colnc * NN;
  v8f acc;
  for (int p = 0; p < 8; ++p) acc[p] = 0.f;
  int kbase = hi ? 2 : 0;
  // main loop: k0 = 0..196, ka+1 <= 199 < 207 always in range
  for (int k0 = 0; k0 < NP - 8; k0 += 4) {
    int ka = k0 + kbase;
    v2f a, b;
    a.x = Ar[ka]     * rm;
    a.y = Ar[ka + 1] * rm;
    b.x = Br[ka]     * cm;
    b.y = Br[ka + 1] * cm;
    acc = __builtin_amdgcn_wmma_f32_16x16x4_f32(false, a, false, b,
                                                (short)0, acc, false, false);
  }
  { // epilogue chunk k0 = 204: ka in {204,206}, ka+1 in {205,207}
    int ka = (NP - 4) + kbase;
    float m0 = (ka     < NN) ? 1.f : 0.f;
    float m1 = (ka + 1 < NN) ? 1.f : 0.f;
    int k0c = ka     < NN ? ka     : NN - 1;
    int k1c = ka + 1 < NN ? ka + 1 : NN - 1;
    v2f a, b;
    a.x = Ar[k0c] * (rm * m0);
    a.y = Ar[k1c] * (rm * m1);
    b.x = Br[k0c] * (cm * m0);
    b.y = Br[k1c] * (cm * m1);
    acc = __builtin_amdgcn_wmma_f32_16x16x4_f32(false, a, false, b,
                                                (short)0, acc, false, false);
  }
  float* Kc = Ks + c * NP2;
  for (int p = 0; p < 8; ++p) {
    int m = (hi ? 8 : 0) + p;
    Kc[(tm*16 + m) * NP + tn*16 + l] = acc[p];
  }
}

// ============ 3) Cholesky of A = Dt_j*Ks_c + sig^2 I (dynamic LDS) =======
// LDS pitch 209 (mod 64 banks -> step 17, coprime) kills the 4-way bank
// conflicts a 208 pitch would have on the column/rank-1 accesses.
__global__ void chol_kernel(const float* __restrict__ Ks,
                            const float* __restrict__ Dt,
                            const float* __restrict__ sigma,
                            float* __restrict__ G) {
  extern __shared__ float A[];          // NP*LDSP floats = 173888 bytes
  int blk = blockIdx.x;                 // c*12 + j
  int c = blk / TT;
  int j = blk % TT;
  float dtj = Dt[c*TT + j];
  float sg  = sigma[c];
  float s2  = sg * sg;
  const float* Kc = Ks + c * NP2;
  for (int idx = threadIdx.x; idx < NP2; idx += blockDim.x) {
    int r = idx / NP, q = idx % NP;
    float v = dtj * Kc[idx];
    if (r == q) v += s2;
    A[r * LDSP + q] = v;
  }
  __syncthreads();
  for (int k = 0; k < NP; ++k) {
    if (threadIdx.x == 0) A[k*LDSP + k] = sqrtf(A[k*LDSP + k]);
    __syncthreads();
    float dk = A[k*LDSP + k];
    for (int i = k + 1 + threadIdx.x; i < NP; i += blockDim.x) A[i*LDSP + k] /= dk;
    __syncthreads();
    for (int i = k + 1 + threadIdx.x; i < NP; i += blockDim.x) {
      float aik = A[i*LDSP + k];
      for (int q = k + 1; q <= i; ++q) A[i*LDSP + q] -= aik * A[q*LDSP + k];
    }
    __syncthreads();
  }
  float* Gm = G + (size_t)blk * NP2;
  for (int idx = threadIdx.x; idx < NP2; idx += blockDim.x)
    Gm[idx] = A[(idx / NP) * LDSP + (idx % NP)];
}

// ============ 4) Z[c][j][i][b] = sum_t (target-mu)[b,i,t] * Ut_c[t][j] ====
__global__ void zmat_kernel(const float* __restrict__ mu,
                            const float* __restrict__ target,
                            const float* __restrict__ Ut,
                            float* __restrict__ Z) {
  __shared__ float U[144];
  int c = blockIdx.y;
  if (threadIdx.x < 144) U[threadIdx.x] = Ut[c*144 + threadIdx.x];
  __syncthreads();
  int b  = threadIdx.x & 63;
  int il = threadIdx.x >> 6;
  int i  = blockIdx.x * 4 + il;         // 0..207
  float r[TT];
  if (i < NN) {
    int base = (b * NN + i) * TT;
    for (int k = 0; k < TT; ++k) r[k] = target[base + k] - mu[base + k];
  } else {
    for (int k = 0; k < TT; ++k) r[k] = 0.f;
  }
  for (int jj = 0; jj < TT; ++jj) {
    float s = 0.f;
    for (int k = 0; k < TT; ++k) s += r[k] * U[k*TT + jj];
    Z[((size_t)(c*TT + jj) * NP + i) * 64 + b] = s;
  }
}

// ============ 5) blocked forward TRSM (64 RHS) + sum of squares ==========
// One WG per (c,j); 4 waves, each owns 16 RHS columns; WMMA panel updates
// (16xK @ Kx16, K chunks of 4); scalar 16x16 diagonal solve. G/Z streams
// are L2-resident; prefetch the next block's tiles while computing.
__global__ void trsm_quad_kernel(const float* __restrict__ G,
                                 const float* __restrict__ Z,
                                 float* __restrict__ quadp) {
  __shared__ float Ylds[NP * 64];       // 53 KB
  __shared__ float accb[16 * 64];
  __shared__ float dtile[16 * 16];
  int blk  = blockIdx.x;                // c*12 + j
  const float* Gm = G + (size_t)blk * NP2;
  const float* Zm = Z + (size_t)blk * NP * 64;
  int tid  = threadIdx.x;
  int lane = tid & 31;
  int wave = tid >> 5;
  int col0 = wave * 16;
  bool hi  = lane >= 16;
  int l    = lane & 15;
  float sumsq = 0.f;
  float yv[16];
  for (int bi = 0; bi < NP / 16; ++bi) {
    int row0 = bi * 16;
    __syncthreads();                    // Y/dtile from previous phase settled
    for (int idx = tid; idx < 256; idx += blockDim.x)
      dtile[idx] = Gm[(row0 + (idx >> 4)) * NP + row0 + (idx & 15)];
    if (bi + 1 < NP / 16) {             // prefetch next block (gfx1250 global_prefetch)
      __builtin_prefetch(&Gm[(row0 + 16 + (tid & 15)) * NP + row0], 0, 1);
      __builtin_prefetch(&Zm[(row0 + 16 + (tid >> 4)) * 64], 0, 1);
    }
    v8f acc;
    for (int p = 0; p < 8; ++p) acc[p] = 0.f;
    for (int kb = 0; kb < bi; ++kb) {
      for (int kc = 0; kc < 4; ++kc) {
        int k0 = kb * 16 + kc * 4 + (hi ? 2 : 0);
        v2f a, b;
        int ar = row0 + l;
        a.x = Gm[ar * NP + k0];
        a.y = Gm[ar * NP + k0 + 1];
        b.x = Ylds[(k0)     * 64 + col0 + l];
        b.y = Ylds[(k0 + 1) * 64 + col0 + l];
        acc = __builtin_amdgcn_wmma_f32_16x16x4_f32(false, a, false, b,
                                                    (short)0, acc, false, false);
      }
    }
    for (int p = 0; p < 8; ++p) {
      int m = (hi ? 8 : 0) + p;
      accb[m * 64 + col0 + l] = acc[p];
    }
    __syncthreads();
    if (tid < 64) {                     // one thread per RHS column (batch)
      int n = tid;
      for (int r = 0; r < 16; ++r) {
        float s = Zm[(row0 + r) * 64 + n] - accb[r * 64 + n];
        for (int q = 0; q < r; ++q) s -= dtile[r * 16 + q] * yv[q];
        float y = s / dtile[r * 16 + r];
        yv[r] = y;
        sumsq += y * y;
        Ylds[(row0 + r) * 64 + n] = y;
      }
    }
  }
  if (tid < 64) quadp[blk * 64 + tid] = sumsq;
}

// ============ 6) masked-MAE block partials (deterministic) ===============
__global__ void mae_part_kernel(const float* __restrict__ mu,
                                const float* __restrict__ target,
                                const float* __restrict__ ut,
                                float* __restrict__ part) {
  __shared__ float sA[256];
  __shared__ float sI[256];
  int tid = threadIdx.x;
  int idx = blockIdx.x * 256 + tid;     // grid sized exactly: 621*256 = NTOT
  float ind = (ut[idx] != 0.f) ? 1.f : 0.f;
  float av  = fabsf(mu[idx] - target[idx]) * ind;
  sA[tid] = av; sI[tid] = ind;
  __syncthreads();
  for (int off = 128; off > 0; off >>= 1) {
    if (tid < off) { sA[tid] += sA[tid + off]; sI[tid] += sI[tid + off]; }
    __syncthreads();
  }
  if (tid == 0) { part[blockIdx.x * 2] = sA[0]; part[blockIdx.x * 2 + 1] = sI[0]; }
}

// ============ 7) logsumexp over components + final combine ===============
__global__ void final_kernel(const float* __restrict__ quadp,
                             const float* __restrict__ Ulog,
                             const float* __restrict__ Vlog,
                             const float* __restrict__ w,
                             const float* __restrict__ part,
                             float* __restrict__ out) {
  __shared__ float s0[64];
  __shared__ float s1[64];
  __shared__ float s2[64];
  int t = threadIdx.x;                  // 64 threads, one per batch
  float ll[CC];
  for (int c = 0; c < CC; ++c) {
    float q = 0.f;
    for (int jj = 0; jj < TT; ++jj) q += quadp[(c*TT + jj) * 64 + t];
    ll[c] = -0.5f * (float)(NN*TT) * LOG2PI - 0.5f * q
            + (float)NN * Vlog[c] + (float)TT * Ulog[c] + logf(w[t*CC + c]);
  }
  float m = ll[0];
  for (int c = 1; c < CC; ++c) m = fmaxf(m, ll[c]);
  float se = 0.f;
  for (int c = 0; c < CC; ++c) se += expf(ll[c] - m);
  s0[t] = -(m + logf(se));
  float pa = 0.f, pi = 0.f;
  for (int i = t; i < 621; i += 64) { pa += part[2*i]; pi += part[2*i + 1]; }
  s1[t] = pa; s2[t] = pi;
  __syncthreads();
  for (int off = 32; off > 0; off >>= 1) {
    if (t < off) { s0[t] += s0[t+off]; s1[t] += s1[t+off]; s2[t] += s2[t+off]; }
    __syncthreads();
  }
  if (t == 0) {
    float nll_mean = s0[0] / (float)BB;
    float mse = (s2[0] > 0.f) ? (s1[0] / s2[0]) : 0.f;
    out[0] = 0.1f * nll_mean + 0.9f * mse;
  }
}

// ============================ host launcher ==============================
extern "C" void kernel_launch(void* const* d_in, const int* in_sizes, int n_in,
                              void* d_out, int out_size, void* d_ws, size_t ws_size,
                              hipStream_t stream) {
  (void)in_sizes; (void)n_in; (void)out_size; (void)ws_size;
  const float* mu     = (const float*)d_in[0];
  const float* target = (const float*)d_in[1];
  const float* utgt   = (const float*)d_in[2];
  const float* w      = (const float*)d_in[3];
  const float* sigma  = (const float*)d_in[4];
  // d_in[5] = R (unused by the reference computation)
  const float* Ls     = (const float*)d_in[6];
  const float* Lt     = (const float*)d_in[7];
  float* W = (float*)d_ws;
  float* maeP  = W + O_MAEP;
  float* Ut    = W + O_UT;
  float* Dt    = W + O_DT;
  float* Ulog  = W + O_ULOG;
  float* Vlog  = W + O_VLOG;
  float* Ks    = W + O_KS;
  float* G     = W + O_G;
  float* Z     = W + O_Z;
  float* quadp = W + O_QUADP;
  float* out   = (float*)d_out;

  kt_eigen_kernel<<<1, 32, 0, stream>>>(Lt, Ls, Ut, Dt, Ulog, Vlog);
  ks_wmma_kernel<<<dim3(169, CC), 32, 0, stream>>>(Ls, Ks);
  chol_kernel<<<CC * TT, 256, NP * LDSP * sizeof(float), stream>>>(Ks, Dt, sigma, G);
  zmat_kernel<<<dim3(NP / 4, CC), 256, 0, stream>>>(mu, target, Ut, Z);
  trsm_quad_kernel<<<CC * TT, 128, 0, stream>>>(G, Z, quadp);
  mae_part_kernel<<<NTOT / 256, 256, 0, stream>>>(mu, target, utgt, maeP);
  final_kernel<<<1, 64, 0, stream>>>(quadp, Ulog, Vlog, w, maeP, out);
}